// DBLoss_40054865002622
// MI455X (gfx1250) — compile-verified
//
#include <hip/hip_runtime.h>
#include <hip/hip_bf16.h>
#include <stdint.h>

// ---------------------------------------------------------------------------
// DBLoss (DBNet) fused single-pass implementation for gfx1250 (MI455X).
//
// Memory floor: 4 x 26.2MB f32 = 104.9MB @ 23.3 TB/s  =>  ~4.5us.
// Reference's per-map argsort (OHEM) is replaced by a one-pass histogram
// selection so the 105MB is streamed exactly once.
//
// CDNA5-specific paths used:
//   * global_load_async_to_lds_b128 + s_wait_asynccnt (double-buffered
//     streaming prefetch into LDS, ASYNCcnt-tracked)
//   * ds_add_f32 LDS atomics for privatized histograms
//   * wave32 shuffle reductions (warpSize == 32)
// ---------------------------------------------------------------------------

#define TPB   256
#define NBLK  640
#define BINS  512

#define WS_SCALARS 16
#define WS_FLOATS  (WS_SCALARS + 4 * BINS)
// ws scalar layout: 0 posCnt_s, 1 posSum_s, 2 negCnt_s,
//                   3 posCnt_b, 4 posSum_b, 5 negCnt_b, 6 l1Sum
// ws[16 .. 16+4*BINS): hist {cnt_s, sum_s, cnt_b, sum_b}

// Ls losses (logits ~ N(0,1), continuous targets): range [-32, 32), width 1/8
#define S_LO    (-32.0f)
#define S_SCALE 8.0f
// Lb losses (logits scaled by R=50): range [-16384, 16384), width 64
#define B_LO    (-16384.0f)
#define B_SCALE (1.0f / 64.0f)

// stable BCE-with-logits: max(x,0) - x*t + log1p(exp(-|x|))
__device__ __forceinline__ float bce_logits(float x, float t) {
  return fmaxf(x, 0.0f) - x * t + __logf(1.0f + __expf(-fabsf(x)));
}

// ---- CDNA5 async global->LDS (inline asm, portable across toolchains) -----
__device__ __forceinline__ void async_ld_b128(uint32_t lds_addr, const void* gaddr) {
  asm volatile("global_load_async_to_lds_b128 %0, %1, off"
               :: "v"(lds_addr), "v"(gaddr)
               : "memory");
}

template <int N>
__device__ __forceinline__ void wait_asynccnt() {
  asm volatile("s_wait_asynccnt %0" :: "i"(N) : "memory");
}
// ---------------------------------------------------------------------------

struct Acc { float pcS, psS, ncS, pcB, psB, ncB, l1; };

__device__ __forceinline__ void process_elem(float x, float t, float h, float g,
                                             Acc& a, float* s_hist) {
  a.l1 += fabsf(h - g);

  // Ls term: pred = proba_map, target = target_proba_map
  float ls = bce_logits(x, t);
  if (t >= 0.0f) {                       // sigmoid(t) >= 0.5 <=> t >= 0
    a.pcS += 1.0f; a.psS += ls;
  } else {
    a.ncS += 1.0f;
    int b = (int)((ls - S_LO) * S_SCALE);
    b = b < 0 ? 0 : (b > BINS - 1 ? BINS - 1 : b);
    atomicAdd(&s_hist[b], 1.0f);         // ds_add_f32
    atomicAdd(&s_hist[BINS + b], ls);
  }

  // Lb term: bin_map = 50*(p - th), target_bin = 50*(tp - tth)
  float xb = 50.0f * (x - h);
  float tb = 50.0f * (t - g);
  float lb = bce_logits(xb, tb);
  if (tb >= 0.0f) {
    a.pcB += 1.0f; a.psB += lb;
  } else {
    a.ncB += 1.0f;
    int b = (int)((lb - B_LO) * B_SCALE);
    b = b < 0 ? 0 : (b > BINS - 1 ? BINS - 1 : b);
    atomicAdd(&s_hist[2 * BINS + b], 1.0f);
    atomicAdd(&s_hist[3 * BINS + b], lb);
  }
}

__global__ void db_zero(float* ws, int n) {
  int i = blockIdx.x * blockDim.x + threadIdx.x;
  if (i < n) ws[i] = 0.0f;
}

__global__ __launch_bounds__(TPB) void db_main(
    const float4* __restrict__ p, const float4* __restrict__ tp,
    const float4* __restrict__ th, const float4* __restrict__ tg,
    float* __restrict__ ws, int n4) {
  __shared__ float  s_hist[4 * BINS];          // 8 KB
  __shared__ float4 s_stage[2][4][TPB];        // 32 KB double buffer

  for (int i = threadIdx.x; i < 4 * BINS; i += TPB) s_hist[i] = 0.0f;
  __syncthreads();

  const int tid    = threadIdx.x;
  const int ntiles = (n4 + TPB - 1) / TPB;
  Acc a = {0.f, 0.f, 0.f, 0.f, 0.f, 0.f, 0.f};

  int tile  = blockIdx.x;
  int stage = 0;
  if (tile < ntiles) {                         // prologue: prefetch tile 0
    int idx = tile * TPB + tid; idx = idx < n4 ? idx : n4 - 1;
    async_ld_b128((uint32_t)(uintptr_t)&s_stage[0][0][tid], p  + idx);
    async_ld_b128((uint32_t)(uintptr_t)&s_stage[0][1][tid], tp + idx);
    async_ld_b128((uint32_t)(uintptr_t)&s_stage[0][2][tid], th + idx);
    async_ld_b128((uint32_t)(uintptr_t)&s_stage[0][3][tid], tg + idx);
  }

  for (; tile < ntiles; tile += gridDim.x) {
    int nxt = tile + gridDim.x;
    if (nxt < ntiles) {                        // prefetch next tile, then
      int idx = nxt * TPB + tid; idx = idx < n4 ? idx : n4 - 1;
      int ns = stage ^ 1;
      async_ld_b128((uint32_t)(uintptr_t)&s_stage[ns][0][tid], p  + idx);
      async_ld_b128((uint32_t)(uintptr_t)&s_stage[ns][1][tid], tp + idx);
      async_ld_b128((uint32_t)(uintptr_t)&s_stage[ns][2][tid], th + idx);
      async_ld_b128((uint32_t)(uintptr_t)&s_stage[ns][3][tid], tg + idx);
      wait_asynccnt<4>();                      // wait: current tile landed
    } else {
      wait_asynccnt<0>();
    }

    int idx = tile * TPB + tid;
    if (idx < n4) {
      float4 vp = s_stage[stage][0][tid];
      float4 vt = s_stage[stage][1][tid];
      float4 vh = s_stage[stage][2][tid];
      float4 vg = s_stage[stage][3][tid];
      process_elem(vp.x, vt.x, vh.x, vg.x, a, s_hist);
      process_elem(vp.y, vt.y, vh.y, vg.y, a, s_hist);
      process_elem(vp.z, vt.z, vh.z, vg.z, a, s_hist);
      process_elem(vp.w, vt.w, vh.w, vg.w, a, s_hist);
    }
    stage ^= 1;
  }

  // wave32 reduction of the 7 scalar accumulators, then global atomics
  float vals[7] = {a.pcS, a.psS, a.ncS, a.pcB, a.psB, a.ncB, a.l1};
#pragma unroll
  for (int k = 0; k < 7; ++k) {
    float v = vals[k];
#pragma unroll
    for (int off = 16; off > 0; off >>= 1) v += __shfl_down(v, off, 32);
    if ((tid & 31) == 0) atomicAdd(&ws[k], v);
  }
  __syncthreads();
  for (int i = tid; i < 4 * BINS; i += TPB)
    atomicAdd(&ws[WS_SCALARS + i], s_hist[i]);
}

// OHEM via histogram top-down scan. Mimics the reference's sentinel trick:
// positive pixels enter the descending sort with value -1.0, so inject
// n_pos entries of -1 into the pool before taking the top n_neg.
__device__ float ohem_loss(float npos, float possum, float navail,
                           float* cnt, float* sum, float lo, float scale) {
  float target = fminf(navail, 3.0f * npos);   // K_OHEM = 3
  int sb = (int)((-1.0f - lo) * scale);
  sb = sb < 0 ? 0 : (sb > BINS - 1 ? BINS - 1 : sb);
  cnt[sb] += npos;
  sum[sb] -= npos;                             // n_pos sentinels of value -1
  float acc = 0.0f, negsum = 0.0f;
  for (int b = BINS - 1; b >= 0; --b) {
    float c = cnt[b];
    if (c <= 0.0f) continue;
    if (acc + c <= target) {
      negsum += sum[b]; acc += c;
    } else {
      float rem = target - acc;
      negsum += sum[b] * (rem / c);            // boundary bin: mean-value take
      break;
    }
  }
  return (possum + negsum) / (npos + target);
}

__global__ void db_final(float* __restrict__ ws, float* __restrict__ out,
                         float n_total) {
  __shared__ float h[4 * BINS];
  for (int i = threadIdx.x; i < 4 * BINS; i += blockDim.x)
    h[i] = ws[WS_SCALARS + i];
  __syncthreads();
  if (threadIdx.x == 0) {
    float Ls = ohem_loss(ws[0], ws[1], ws[2], h,            h + BINS,     S_LO, S_SCALE);
    float Lb = ohem_loss(ws[3], ws[4], ws[5], h + 2 * BINS, h + 3 * BINS, B_LO, B_SCALE);
    float Lt = ws[6] / n_total;
    out[0] = Ls + 1.0f * Lb + 10.0f * Lt;      // ALPHA=1, BETA=10
  }
}

extern "C" void kernel_launch(void* const* d_in, const int* in_sizes, int n_in,
                              void* d_out, int out_size, void* d_ws, size_t ws_size,
                              hipStream_t stream) {
  (void)n_in; (void)out_size; (void)ws_size;
  const float4* p  = (const float4*)d_in[0];   // proba_map
  const float4* tp = (const float4*)d_in[1];   // target_proba_map
  const float4* th = (const float4*)d_in[2];   // thresh_map
  const float4* tg = (const float4*)d_in[3];   // target_thresh_map
  float* ws = (float*)d_ws;
  int n  = in_sizes[0];                        // 16*640*640, divisible by 4
  int n4 = n >> 2;

  db_zero<<<(WS_FLOATS + 255) / 256, 256, 0, stream>>>(ws, WS_FLOATS);
  db_main<<<NBLK, TPB, 0, stream>>>(p, tp, th, tg, ws, n4);
  db_final<<<1, 256, 0, stream>>>(ws, (float*)d_out, (float)n);
}